// SoftRegionOperator_8959301779651
// MI455X (gfx1250) — compile-verified
//
#include <hip/hip_runtime.h>
#include <math.h>

typedef float v2f __attribute__((ext_vector_type(2)));
typedef float v8f __attribute__((ext_vector_type(8)));

#define WMMA_F32(a, b, c) \
  __builtin_amdgcn_wmma_f32_16x16x4_f32(false, (a), false, (b), (short)0, (c), false, false)

__device__ __forceinline__ v2f ld2(const float* p) {
  return *reinterpret_cast<const v2f*>(p);
}

template <int ACT>
__device__ __forceinline__ float actf(float v) {
  if constexpr (ACT == 1) {
    return 0.5f * v * (1.0f + erff(v * 0.70710678118654752f));  // exact gelu
  } else if constexpr (ACT == 2) {
    return fmaxf(v, 0.0f);
  } else {
    return v;
  }
}

// K-loop helpers: one A fragment feeds two output-column WMMAs (1.5 loads/wmma)
template <int K>
__device__ __forceinline__ void gemm_pair(const float* __restrict__ aP,
                                          const float* __restrict__ b0,
                                          const float* __restrict__ b1,
                                          v8f& c0, v8f& c1) {
#pragma unroll 4
  for (int k0 = 0; k0 < K; k0 += 4) {
    const v2f a = ld2(aP + k0);
    c0 = WMMA_F32(a, ld2(b0 + k0), c0);
    c1 = WMMA_F32(a, ld2(b1 + k0), c1);
  }
}

template <int K>
__device__ __forceinline__ void gemm_one(const float* __restrict__ aP,
                                         const float* __restrict__ b0, v8f& c0) {
#pragma unroll 4
  for (int k0 = 0; k0 < K; k0 += 4) {
    c0 = WMMA_F32(ld2(aP + k0), ld2(b0 + k0), c0);
  }
}

// ---------------------------------------------------------------------------
// Stage A: G[o][k] = sum_j K[o][j] * e^{-2*pi*i*j*k/256},  K = kr + i*ki
// ---------------------------------------------------------------------------
__global__ __launch_bounds__(256) void sro_build_G(const float* __restrict__ kr,
                                                   const float* __restrict__ ki,
                                                   float* __restrict__ Gr,
                                                   float* __restrict__ Gi) {
  __shared__ float ct[256], st[256], arow[256], brow[256];
  const int k = threadIdx.x;
  const int o = blockIdx.x;
  const float ang = 0.024543692606170259f * (float)k;  // 2*pi/256 * k
  ct[k] = cosf(ang);
  st[k] = sinf(ang);
  arow[k] = kr[o * 256 + k];
  brow[k] = ki[o * 256 + k];
  __syncthreads();
  float gr = 0.0f, gi = 0.0f;
  for (int j = 0; j < 256; ++j) {
    const int t = (j * k) & 255;
    const float c = ct[t], s = st[t];
    const float a = arow[j], b = brow[j];
    gr = fmaf(a, c, fmaf(b, s, gr));
    gi = fmaf(b, c, fmaf(-a, s, gi));
  }
  Gr[o * 256 + k] = gr;
  Gi[o * 256 + k] = gi;
}

// ---------------------------------------------------------------------------
// Stage B: M[n][k] = (1/256) * sum_o ( cos(2pi*n*o/256)*Gr[o][k]
//                                     - sin(2pi*n*o/256)*Gi[o][k] )
// y0 = x @ M^T  (M stored like a weight matrix [out][in])
// ---------------------------------------------------------------------------
__global__ __launch_bounds__(256) void sro_build_M(const float* __restrict__ Gr,
                                                   const float* __restrict__ Gi,
                                                   float* __restrict__ M) {
  __shared__ float ct[256], st[256];
  const int k = threadIdx.x;
  const int n = blockIdx.x;
  const float ang = 0.024543692606170259f * (float)k;
  ct[k] = cosf(ang);
  st[k] = sinf(ang);
  __syncthreads();
  float acc = 0.0f;
  for (int o = 0; o < 256; ++o) {
    const int t = (n * o) & 255;
    acc = fmaf(ct[t], Gr[o * 256 + k], fmaf(-st[t], Gi[o * 256 + k], acc));
  }
  M[n * 256 + k] = acc * 0.00390625f;  // 1/256
}

// ---------------------------------------------------------------------------
// Bucketing: compact row indices per region
// ---------------------------------------------------------------------------
__global__ void sro_zero_counts(int* __restrict__ cnt) {
  if (threadIdx.x < 8) cnt[threadIdx.x] = 0;
}

__global__ void sro_scatter_idx(const int* __restrict__ mask, int* __restrict__ cnt,
                                int* __restrict__ idx, int n) {
  const int i = blockIdx.x * 256 + threadIdx.x;
  if (i < n) {
    const int m = mask[i];
    const int p = atomicAdd(&cnt[m], 1);
    idx[m * n + p] = i;
  }
}

// ---------------------------------------------------------------------------
// Routed branch GEMM. One wave owns one 16-row gathered tile.
// H == 0  : single GEMM with Wm (the FFT-equivalent matrix), no bias/act.
// H >  0  : 3-layer MLP, hidden dim H (multiple of 16), activation ACT.
// WMMA: V_WMMA_F32_16X16X4_F32, fp32 A/B/C.
// ---------------------------------------------------------------------------
template <int H, int ACT>
__global__ __launch_bounds__(128) void sro_branch(
    const float* __restrict__ x, const float* __restrict__ W1,
    const float* __restrict__ B1, const float* __restrict__ W2,
    const float* __restrict__ B2, const float* __restrict__ W3,
    const float* __restrict__ B3, const float* __restrict__ Wm,
    const int* __restrict__ idxlist, const int* __restrict__ cntp,
    float* __restrict__ out) {
  constexpr int WAVES = 4;
  constexpr int HS = 68;  // padded LDS row stride (floats) -> conflict-free
  constexpr int HT = (H > 0) ? (H / 16) : 0;
  __shared__ float Hs[(H > 0) ? WAVES * 16 * HS : 1];
  __shared__ float H2s[(H > 0) ? WAVES * 16 * HS : 1];

  const int count = *cntp;
  const int ntiles = (count + 15) >> 4;
  if ((int)blockIdx.x * WAVES >= ntiles) return;  // block-uniform early out

  const int waveId = threadIdx.x >> 5;
  const int lane = threadIdx.x & 31;
  const int tile = blockIdx.x * WAVES + waveId;
  const bool active = (tile < ntiles);  // uniform per wave
  const int l = lane & 15;
  const int h = lane >> 4;

  int gi = 0, ok = 0;
  if (active && lane < 16) {
    const int raw = tile * 16 + lane;
    ok = (raw < count) ? 1 : 0;
    gi = ok ? idxlist[raw] : idxlist[count - 1];
  }
  const int rowA = __shfl(gi, l, 32);
  const float* xA = x + (size_t)rowA * 256 + 2 * h;

  // Hoisted per-lane scatter info: D element (m = r + 8h, n = l).
  // Output offsets fit in 31 bits (131072*256 < 2^31).
  int okm8[8];
  int outoff[8];
#pragma unroll
  for (int r = 0; r < 8; ++r) {
    const int m = r + 8 * h;
    okm8[r] = __shfl(ok, m, 32);
    outoff[r] = __shfl(gi, m, 32) * 256 + l;
  }

  if constexpr (H == 0) {
    if (active) {
      for (int ot = 0; ot < 16; ot += 2) {
        v8f c0 = {0.f, 0.f, 0.f, 0.f, 0.f, 0.f, 0.f, 0.f};
        v8f c1 = {0.f, 0.f, 0.f, 0.f, 0.f, 0.f, 0.f, 0.f};
        const float* wB0 = Wm + (size_t)(ot * 16 + l) * 256 + 2 * h;
        gemm_pair<256>(xA, wB0, wB0 + 16 * 256, c0, c1);
#pragma unroll
        for (int r = 0; r < 8; ++r) {
          if (okm8[r]) {
            float* o = out + outoff[r] + ot * 16;
            o[0] = c0[r];
            o[16] = c1[r];
          }
        }
      }
    }
    return;  // no barriers in this path
  } else {
    float* hs = &Hs[waveId * 16 * HS];
    float* h2s = &H2s[waveId * 16 * HS];

    // ---- stage 1: h1 = act(x @ W1^T + b1), K=256, OUT=H ----
    if (active) {
      int ot = 0;
      for (; ot + 1 < HT; ot += 2) {
        v8f c0 = {0.f, 0.f, 0.f, 0.f, 0.f, 0.f, 0.f, 0.f};
        v8f c1 = {0.f, 0.f, 0.f, 0.f, 0.f, 0.f, 0.f, 0.f};
        const float* wB0 = W1 + (size_t)(ot * 16 + l) * 256 + 2 * h;
        gemm_pair<256>(xA, wB0, wB0 + 16 * 256, c0, c1);
        const float b0 = B1[ot * 16 + l];
        const float b1 = B1[ot * 16 + 16 + l];
#pragma unroll
        for (int r = 0; r < 8; ++r) {
          float* hrow = hs + (r + 8 * h) * HS + ot * 16 + l;
          hrow[0] = actf<ACT>(c0[r] + b0);
          hrow[16] = actf<ACT>(c1[r] + b1);
        }
      }
      if (ot < HT) {  // odd tail (H=48)
        v8f c0 = {0.f, 0.f, 0.f, 0.f, 0.f, 0.f, 0.f, 0.f};
        const float* wB0 = W1 + (size_t)(ot * 16 + l) * 256 + 2 * h;
        gemm_one<256>(xA, wB0, c0);
        const float b0 = B1[ot * 16 + l];
#pragma unroll
        for (int r = 0; r < 8; ++r)
          hs[(r + 8 * h) * HS + ot * 16 + l] = actf<ACT>(c0[r] + b0);
      }
    }
    __syncthreads();

    // ---- stage 2: h2 = act(h1 @ W2^T + b2), K=H, OUT=H ----
    if (active) {
      const float* aP = hs + l * HS + 2 * h;
      int ot = 0;
      for (; ot + 1 < HT; ot += 2) {
        v8f c0 = {0.f, 0.f, 0.f, 0.f, 0.f, 0.f, 0.f, 0.f};
        v8f c1 = {0.f, 0.f, 0.f, 0.f, 0.f, 0.f, 0.f, 0.f};
        const float* wB0 = W2 + (size_t)(ot * 16 + l) * H + 2 * h;
        gemm_pair<H>(aP, wB0, wB0 + 16 * H, c0, c1);
        const float b0 = B2[ot * 16 + l];
        const float b1 = B2[ot * 16 + 16 + l];
#pragma unroll
        for (int r = 0; r < 8; ++r) {
          float* hrow = h2s + (r + 8 * h) * HS + ot * 16 + l;
          hrow[0] = actf<ACT>(c0[r] + b0);
          hrow[16] = actf<ACT>(c1[r] + b1);
        }
      }
      if (ot < HT) {
        v8f c0 = {0.f, 0.f, 0.f, 0.f, 0.f, 0.f, 0.f, 0.f};
        const float* wB0 = W2 + (size_t)(ot * 16 + l) * H + 2 * h;
        gemm_one<H>(aP, wB0, c0);
        const float b0 = B2[ot * 16 + l];
#pragma unroll
        for (int r = 0; r < 8; ++r)
          h2s[(r + 8 * h) * HS + ot * 16 + l] = actf<ACT>(c0[r] + b0);
      }
    }
    __syncthreads();

    // ---- stage 3: y = h2 @ W3^T + b3, K=H, OUT=256, scatter rows ----
    if (active) {
      const float* aP = h2s + l * HS + 2 * h;
      for (int ot = 0; ot < 16; ot += 2) {
        v8f c0 = {0.f, 0.f, 0.f, 0.f, 0.f, 0.f, 0.f, 0.f};
        v8f c1 = {0.f, 0.f, 0.f, 0.f, 0.f, 0.f, 0.f, 0.f};
        const float* wB0 = W3 + (size_t)(ot * 16 + l) * H + 2 * h;
        gemm_pair<H>(aP, wB0, wB0 + 16 * H, c0, c1);
        const float b0 = B3[ot * 16 + l];
        const float b1 = B3[ot * 16 + 16 + l];
#pragma unroll
        for (int r = 0; r < 8; ++r) {
          if (okm8[r]) {
            float* o = out + outoff[r] + ot * 16;
            o[0] = c0[r] + b0;
            o[16] = c1[r] + b1;
          }
        }
      }
    }
  }
}

// ---------------------------------------------------------------------------
extern "C" void kernel_launch(void* const* d_in, const int* in_sizes, int n_in,
                              void* d_out, int out_size, void* d_ws, size_t ws_size,
                              hipStream_t stream) {
  const float* x = (const float*)d_in[0];
  const int* mask = (const int*)d_in[1];
  const float* fft_kr = (const float*)d_in[2];
  const float* fft_ki = (const float*)d_in[3];
  const float* fno_w1 = (const float*)d_in[4];
  const float* fno_b1 = (const float*)d_in[5];
  const float* fno_w2 = (const float*)d_in[6];
  const float* fno_b2 = (const float*)d_in[7];
  const float* fno_w3 = (const float*)d_in[8];
  const float* fno_b3 = (const float*)d_in[9];
  const float* psd_w1 = (const float*)d_in[10];
  const float* psd_b1 = (const float*)d_in[11];
  const float* psd_w2 = (const float*)d_in[12];
  const float* psd_b2 = (const float*)d_in[13];
  const float* psd_w3 = (const float*)d_in[14];
  const float* psd_b3 = (const float*)d_in[15];
  const float* m3_w1 = (const float*)d_in[16];
  const float* m3_b1 = (const float*)d_in[17];
  const float* m3_w2 = (const float*)d_in[18];
  const float* m3_b2 = (const float*)d_in[19];
  const float* m3_w3 = (const float*)d_in[20];
  const float* m3_b3 = (const float*)d_in[21];
  const float* m4_w1 = (const float*)d_in[22];
  const float* m4_b1 = (const float*)d_in[23];
  const float* m4_w2 = (const float*)d_in[24];
  const float* m4_b2 = (const float*)d_in[25];
  const float* m4_w3 = (const float*)d_in[26];
  const float* m4_b3 = (const float*)d_in[27];
  float* out = (float*)d_out;

  const int n = in_sizes[0] / 256;  // 131072 rows

  char* ws = (char*)d_ws;
  float* Mws = (float*)(ws + 0);        // 256*256 f32
  float* Gr = (float*)(ws + 262144);    // 256*256 f32
  float* Gi = (float*)(ws + 524288);    // 256*256 f32
  int* cnt = (int*)(ws + 786432);       // 5 counters (padded)
  int* idx = (int*)(ws + 786688);       // 5 * n indices

  // FFT-equivalent matrix M (tiny)
  sro_build_G<<<256, 256, 0, stream>>>(fft_kr, fft_ki, Gr, Gi);
  sro_build_M<<<256, 256, 0, stream>>>(Gr, Gi, Mws);

  // bucket rows by region
  sro_zero_counts<<<1, 32, 0, stream>>>(cnt);
  sro_scatter_idx<<<(n + 255) / 256, 256, 0, stream>>>(mask, cnt, idx, n);

  // routed branch GEMMs: grid covers worst case (all rows in one bucket)
  const int blocks = (((n + 15) / 16) + 3) / 4;
  sro_branch<0, 0><<<blocks, 128, 0, stream>>>(
      x, nullptr, nullptr, nullptr, nullptr, nullptr, nullptr, Mws,
      idx + 0 * n, cnt + 0, out);
  sro_branch<48, 1><<<blocks, 128, 0, stream>>>(
      x, fno_w1, fno_b1, fno_w2, fno_b2, fno_w3, fno_b3, nullptr,
      idx + 1 * n, cnt + 1, out);
  sro_branch<64, 1><<<blocks, 128, 0, stream>>>(
      x, psd_w1, psd_b1, psd_w2, psd_b2, psd_w3, psd_b3, nullptr,
      idx + 2 * n, cnt + 2, out);
  sro_branch<32, 2><<<blocks, 128, 0, stream>>>(
      x, m3_w1, m3_b1, m3_w2, m3_b2, m3_w3, m3_b3, nullptr,
      idx + 3 * n, cnt + 3, out);
  sro_branch<32, 2><<<blocks, 128, 0, stream>>>(
      x, m4_w1, m4_b1, m4_w2, m4_b2, m4_w3, m4_b3, nullptr,
      idx + 4 * n, cnt + 4, out);
}